// EdgeAggregatorGATED_16595753632163
// MI455X (gfx1250) — compile-verified
//
#include <hip/hip_runtime.h>
#include <hip/hip_bf16.h>

// ResGatedGraphConv, MI455X (gfx1250, wave32).
// N=50000 nodes (D=64), E=800000 edges (ED=32).
// Phase 1: node projections via V_WMMA_F32_16X16X4_F32 (fp32 matrix pipe).
// Phase 2: fused edge GEMM + gather + sigmoid gate + atomic scatter-add.

#define N_NODES 50000
#define N_EDGES 800000
#define DN 64
#define DE 32

typedef __attribute__((ext_vector_type(2))) float v2f;
typedef __attribute__((ext_vector_type(8))) float v8f;

__device__ __forceinline__ v8f wmma_f32_16x16x4(v2f a, v2f b, v8f c) {
  // 8 args: (neg_a, A, neg_b, B, c_mod, C, reuse_a, reuse_b)
  return __builtin_amdgcn_wmma_f32_16x16x4_f32(
      false, a, false, b, (short)0, c, false, false);
}

__device__ __forceinline__ void atomic_add_f32(float* p, float v) {
  // relaxed, agent scope, result unused -> atomic-without-return (STOREcnt)
  __hip_atomic_fetch_add(p, v, __ATOMIC_RELAXED, __HIP_MEMORY_SCOPE_AGENT);
}

// ---------------------------------------------------------------------------
// Kernel 1: node projections.
//   grid = (ceil(3125/4), 2), block = 256 (8 waves).
//   blockIdx.y selects weight pair {Wk,Wq} or {Wv,Wskip}; 32KB LDS holds the
//   two [64,64] node-part weights transposed as wT[col][k] (b64 LDS reads).
//   wave w: matrix = 2*blockIdx.y + (w&1), row tile = blockIdx.x*4 + (w>>1).
// ---------------------------------------------------------------------------
__global__ __launch_bounds__(256) void node_proj_kernel(
    const float* __restrict__ x,
    const float* __restrict__ Wk, const float* __restrict__ Wq,
    const float* __restrict__ Wv, const float* __restrict__ Wskip,
    const float* __restrict__ bias,
    float* __restrict__ kx, float* __restrict__ qx, float* __restrict__ vx,
    float* __restrict__ out) {
  __shared__ float wT[2 * DN * DN];  // 32 KB: [mat_local][col][k]

  const int t = threadIdx.x;
  const float* W0 = (blockIdx.y == 0) ? Wk : Wv;
  const float* W1 = (blockIdx.y == 0) ? Wq : Wskip;
#pragma unroll
  for (int i = 0; i < 16; ++i) {
    const int idx = t + i * 256;    // 0..4095
    const int c = idx & 63;         // coalesced on global side
    const int k = idx >> 6;
    wT[c * DN + k] = W0[k * DN + c];
    wT[DN * DN + c * DN + k] = W1[k * DN + c];
  }
  __syncthreads();

  const int wave = t >> 5;
  const int lane = t & 31;
  const int half = lane >> 4;        // K-half selector (A/B frag layout)
  const int ln = lane & 15;          // row (A) / col (B,C) within tile
  const int mat_local = wave & 1;
  const int mat = (blockIdx.y << 1) | mat_local;
  const int m0 = (blockIdx.x * 4 + (wave >> 1)) * 16;
  if (m0 >= N_NODES) return;         // wave-uniform guard (EXEC stays all-1)

  const float* __restrict__ wTm = &wT[mat_local * DN * DN];

  v8f acc[4] = {v8f{}, v8f{}, v8f{}, v8f{}};
  for (int kk = 0; kk < DN; kk += 4) {
    const int kb = kk + 2 * half;
    // A fragment: rows m0+ln, K = kb, kb+1 (contiguous fp32 pair)
    const float* ar = x + (size_t)(m0 + ln) * DN + kb;
    v2f a;
    a.x = ar[0];
    a.y = ar[1];
#pragma unroll
    for (int n = 0; n < 4; ++n) {
      // B fragment from LDS: wT[col][kb], wT[col][kb+1] -> single b64 read
      const float* bp = wTm + (n * 16 + ln) * DN + kb;
      v2f b;
      b.x = bp[0];
      b.y = bp[1];
      acc[n] = wmma_f32_16x16x4(a, b, acc[n]);
    }
  }

  float* __restrict__ dstp = (mat == 0) ? kx : (mat == 1) ? qx
                            : (mat == 2) ? vx : out;
#pragma unroll
  for (int n = 0; n < 4; ++n) {
    const int col = n * 16 + ln;
    const float bcol = (mat == 3) ? bias[col] : 0.0f;
#pragma unroll
    for (int v = 0; v < 8; ++v) {
      const int row = m0 + v + 8 * half;   // C layout: VGPR v, half selects M+8
      dstp[(size_t)row * DN + col] = acc[n][v] + bcol;
    }
  }
}

// ---------------------------------------------------------------------------
// Kernel 2: fused edge GEMM + gather + gate + scatter.
//   grid = 6250, block = 256 (8 waves); wave handles one 16-edge tile.
//   24KB LDS: 3 edge-part weights [32,64] transposed as wT[col][k].
//   Per 16-col tile: 3 WMMA chains (k/q/v) over K=32, then gather
//   kx[dst]/qx[src]/vx[src] (L2-resident), sigmoid gate, fp32 atomic add.
// ---------------------------------------------------------------------------
__global__ __launch_bounds__(256) void edge_msg_kernel(
    const float* __restrict__ edge_attr, const int* __restrict__ edge_index,
    const float* __restrict__ Wk, const float* __restrict__ Wq,
    const float* __restrict__ Wv,
    const float* __restrict__ bk, const float* __restrict__ bq,
    const float* __restrict__ bv,
    const float* __restrict__ kx, const float* __restrict__ qx,
    const float* __restrict__ vx,
    float* __restrict__ out) {
  __shared__ float wT[3 * DN * DE];  // 24 KB: [mat][col][k]

  const int t = threadIdx.x;
  const float* const Wsrc[3] = {Wk, Wq, Wv};
#pragma unroll
  for (int m = 0; m < 3; ++m) {
    const float* W = Wsrc[m] + DN * DN;  // edge-part rows D..D+ED-1
#pragma unroll
    for (int i = 0; i < 8; ++i) {
      const int idx = t + i * 256;  // 0..2047
      const int c = idx & 63;
      const int k = idx >> 6;       // 0..31
      wT[m * (DN * DE) + c * DE + k] = W[k * DN + c];
    }
  }
  __syncthreads();

  const int wave = t >> 5;
  const int lane = t & 31;
  const int half = lane >> 4;
  const int ln = lane & 15;
  const int e0 = (blockIdx.x * 8 + wave) * 16;  // 50000 tiles exactly

  // Edge endpoints for the 8 C-rows this lane owns (e = e0 + v + 8*half).
  int se[8], de[8];
#pragma unroll
  for (int v = 0; v < 8; ++v) {
    const int e = e0 + v + 8 * half;
    se[v] = edge_index[e];            // src (j)
    de[v] = edge_index[N_EDGES + e];  // dst (i)
  }

  const float* __restrict__ wTk = wT;
  const float* __restrict__ wTq = wT + DN * DE;
  const float* __restrict__ wTv = wT + 2 * DN * DE;

  for (int n = 0; n < 4; ++n) {      // 4 column tiles of 16
    v8f ak = {}, aq = {}, av = {};
#pragma unroll
    for (int kk = 0; kk < DE; kk += 4) {
      const int kb = kk + 2 * half;
      const float* ar = edge_attr + (size_t)(e0 + ln) * DE + kb;
      v2f a;
      a.x = ar[0];
      a.y = ar[1];
      const int wi = (n * 16 + ln) * DE + kb;
      v2f b0, b1, b2;
      b0.x = wTk[wi]; b0.y = wTk[wi + 1];
      b1.x = wTq[wi]; b1.y = wTq[wi + 1];
      b2.x = wTv[wi]; b2.y = wTv[wi + 1];
      ak = wmma_f32_16x16x4(a, b0, ak);
      aq = wmma_f32_16x16x4(a, b1, aq);
      av = wmma_f32_16x16x4(a, b2, av);
    }

    const int col = n * 16 + ln;
    const float bkc = bk[col];
    const float bqc = bq[col];
    const float bvc = bv[col];
#pragma unroll
    for (int v = 0; v < 8; ++v) {
      const float ki = ak[v] + kx[(size_t)de[v] * DN + col] + bkc;
      const float qj = aq[v] + qx[(size_t)se[v] * DN + col] + bqc;
      const float vj = av[v] + vx[(size_t)se[v] * DN + col] + bvc;
      const float g = 1.0f / (1.0f + __expf(-(ki + qj)));
      atomic_add_f32(out + (size_t)de[v] * DN + col, g * vj);
    }
  }
}

// ---------------------------------------------------------------------------
extern "C" void kernel_launch(void* const* d_in, const int* in_sizes, int n_in,
                              void* d_out, int out_size, void* d_ws,
                              size_t ws_size, hipStream_t stream) {
  const float* x = (const float*)d_in[0];
  const int* edge_index = (const int*)d_in[1];
  const float* edge_attr = (const float*)d_in[2];
  const float* Wk = (const float*)d_in[3];
  const float* bk = (const float*)d_in[4];
  const float* Wq = (const float*)d_in[5];
  const float* bq = (const float*)d_in[6];
  const float* Wv = (const float*)d_in[7];
  const float* bv = (const float*)d_in[8];
  const float* Wskip = (const float*)d_in[9];
  const float* bias = (const float*)d_in[10];
  float* out = (float*)d_out;

  float* kx = (float*)d_ws;                    // [N, 64]
  float* qx = kx + (size_t)N_NODES * DN;       // [N, 64]
  float* vx = qx + (size_t)N_NODES * DN;       // [N, 64]

  // Phase 1: node projections (+ skip path + bias -> out, re-inits d_out).
  const int node_tiles = (N_NODES + 15) / 16;              // 3125
  dim3 g1((node_tiles + 3) / 4, 2);
  node_proj_kernel<<<g1, 256, 0, stream>>>(x, Wk, Wq, Wv, Wskip, bias,
                                           kx, qx, vx, out);

  // Phase 2: fused edge message + scatter (800000/16/8 = 6250 blocks exact).
  edge_msg_kernel<<<N_EDGES / (16 * 8), 256, 0, stream>>>(
      edge_attr, edge_index, Wk, Wq, Wv, bk, bq, bv, kx, qx, vx, out);
}